// WeightedGCNConv_40321152974901
// MI455X (gfx1250) — compile-verified
//
#include <hip/hip_runtime.h>

#define C 128
#define LN_EPS 1e-5f

typedef __attribute__((ext_vector_type(2))) float v2f;
typedef __attribute__((ext_vector_type(8))) float v8f;

// ---------------- degree / normalization ----------------

__global__ void k_init_deg(float* __restrict__ deg, int n) {
  int i = blockIdx.x * blockDim.x + threadIdx.x;
  if (i < n) deg[i] = 1.0f;            // self-loop weight baked in
}

__global__ void k_deg_acc(const int* __restrict__ dst, const float* __restrict__ ew,
                          float* __restrict__ deg, int E) {
  int e = blockIdx.x * blockDim.x + threadIdx.x;
  if (e < E) atomicAdd(&deg[dst[e]], ew[e]);
}

__global__ void k_dinv(const float* __restrict__ deg, float* __restrict__ d, int n) {
  int i = blockIdx.x * blockDim.x + threadIdx.x;
  if (i < n) d[i] = fminf(rsqrtf(deg[i]), 10000.0f);
}

// agg initialized with self-loop contribution: h[i] * d[i]^2
__global__ void k_self(const float* __restrict__ h, const float* __restrict__ d,
                       float* __restrict__ agg, int n) {
  int t = blockIdx.x * blockDim.x + threadIdx.x;
  int i = t >> 5, q = t & 31;          // 32 float4 per row
  if (i < n) {
    float di2 = d[i] * d[i];
    float4 v = ((const float4*)(h + (size_t)i * C))[q];
    v.x *= di2; v.y *= di2; v.z *= di2; v.w *= di2;
    ((float4*)(agg + (size_t)i * C))[q] = v;
  }
}

// ---------------- edge scatter: one wave32 per edge ----------------

__global__ void k_scatter(const float* __restrict__ h, const float* __restrict__ d,
                          const int* __restrict__ src, const int* __restrict__ dst,
                          const float* __restrict__ ew, float* __restrict__ agg, int E) {
  int gt = blockIdx.x * blockDim.x + threadIdx.x;
  int e = gt >> 5;                     // wave id == edge id
  int lane = gt & 31;
  if (e >= E) return;
  int s = src[e], t = dst[e];
  float nrm = d[s] * ew[e] * d[t];
  float4 v = ((const float4*)(h + (size_t)s * C))[lane];
  float* ap = agg + (size_t)t * C + 4 * lane;
  atomicAdd(ap + 0, v.x * nrm);
  atomicAdd(ap + 1, v.y * nrm);
  atomicAdd(ap + 2, v.z * nrm);
  atomicAdd(ap + 3, v.w * nrm);
}

// ---------------- fused fp32-WMMA GEMM + bias + residual + LN + GELU ----------------
// Block: 256 threads = 8 waves. Block computes a 16-row x 128-col x-tile.
// A tile (16x128) is staged once into LDS (padded stride 132 -> conflict-free,
// 8B-aligned b64 fragment reads); every wave feeds its WMMA A fragments from LDS.
// Wave w owns output columns [16w, 16w+16); K-loop of 32 V_WMMA_F32_16X16X4_F32.
// The same LDS buffer is reused (after a barrier) as the x-tile for LayerNorm.

#define TSTRIDE 132   // padded LDS stride (floats)

__global__ __launch_bounds__(256) void k_gemm_ln_gelu(
    const float* __restrict__ agg, const float* __restrict__ h,
    const float* __restrict__ Wm,  const float* __restrict__ bias,
    const float* __restrict__ gamma, const float* __restrict__ beta,
    float* __restrict__ out, int n) {
  __shared__ float smem[16 * TSTRIDE];

  int tid = threadIdx.x;
  int wave = tid >> 5;
  int lane = tid & 31;
  int lm   = lane & 15;
  int kh   = (lane >> 4) << 1;         // 0 for lanes 0-15, 2 for lanes 16-31
  int rowBase = blockIdx.x * 16;
  int colBase = wave * 16;

  // ---- stage A tile (16 x 128) into LDS: 512 float4, 2 per thread ----
  for (int it = tid; it < 16 * 32; it += 256) {
    int rr = it >> 5;                  // tile row 0..15
    int qq = it & 31;                  // float4 index within row
    int row = rowBase + rr;
    if (row >= n) row = n - 1;         // clamp (keeps EXEC full; tail-safe)
    float4 v = ((const float4*)(agg + (size_t)row * C))[qq];
    *(float4*)&smem[rr * TSTRIDE + qq * 4] = v;
  }
  __syncthreads();

  const float* aptr = smem + lm * TSTRIDE + kh;                 // A[m][k..k+1] from LDS
  const float* bptr = Wm + (size_t)(colBase + lm) * C + kh;     // B[k][c] = W[c][k]

  v8f acc = {0.f, 0.f, 0.f, 0.f, 0.f, 0.f, 0.f, 0.f};
#pragma unroll
  for (int k = 0; k < C; k += 4) {
    v2f a = *(const v2f*)(aptr + k);   // ds_load_b64, bank-conflict-free
    v2f b = *(const v2f*)(bptr + k);   // global, L0/L2-resident W
    acc = __builtin_amdgcn_wmma_f32_16x16x4_f32(
        /*neg_a=*/false, a, /*neg_b=*/false, b,
        /*c_mod=*/(short)0, acc, /*reuse_a=*/false, /*reuse_b=*/false);
  }
  __syncthreads();                     // all waves done reading the A tile

  // ---- x = acc + bias + residual -> reuse smem as the x tile ----
  int col = colBase + lm;
  float bc = bias[col];
#pragma unroll
  for (int g = 0; g < 8; ++g) {
    int r = g + ((lane < 16) ? 0 : 8); // D layout: VGPR g -> row g / g+8
    int row = rowBase + r;
    if (row < n)
      smem[r * TSTRIDE + col] = acc[g] + bc + h[(size_t)row * C + col];
  }
  __syncthreads();

  // ---- LayerNorm + exact GELU: 16 threads per row, 8 contiguous cols each ----
  int r   = tid >> 4;
  int sub = tid & 15;
  float4 x0 = *(const float4*)&smem[r * TSTRIDE + sub * 8];      // 16B aligned
  float4 x1 = *(const float4*)&smem[r * TSTRIDE + sub * 8 + 4];
  float s  = x0.x + x0.y + x0.z + x0.w + x1.x + x1.y + x1.z + x1.w;
  float sq = x0.x*x0.x + x0.y*x0.y + x0.z*x0.z + x0.w*x0.w
           + x1.x*x1.x + x1.y*x1.y + x1.z*x1.z + x1.w*x1.w;
#pragma unroll
  for (int o = 8; o > 0; o >>= 1) {    // reduce within 16-lane groups
    s  += __shfl_xor(s,  o, 16);
    sq += __shfl_xor(sq, o, 16);
  }
  float mean = s * (1.0f / C);
  float var  = sq * (1.0f / C) - mean * mean;
  float rstd = rsqrtf(var + LN_EPS);

  int row = rowBase + r;
  if (row < n) {
    float4 g0 = ((const float4*)gamma)[sub * 2];
    float4 g1 = ((const float4*)gamma)[sub * 2 + 1];
    float4 b0 = ((const float4*)beta)[sub * 2];
    float4 b1 = ((const float4*)beta)[sub * 2 + 1];
    float xi[8] = {x0.x, x0.y, x0.z, x0.w, x1.x, x1.y, x1.z, x1.w};
    float gi[8] = {g0.x, g0.y, g0.z, g0.w, g1.x, g1.y, g1.z, g1.w};
    float bi[8] = {b0.x, b0.y, b0.z, b0.w, b1.x, b1.y, b1.z, b1.w};
    float yo[8];
#pragma unroll
    for (int j = 0; j < 8; ++j) {
      float y = gi[j] * (xi[j] - mean) * rstd + bi[j];
      yo[j] = 0.5f * y * (1.0f + erff(y * 0.70710678118654752f));
    }
    float4* op = (float4*)(out + (size_t)row * C + sub * 8);
    op[0] = make_float4(yo[0], yo[1], yo[2], yo[3]);   // two b128 stores,
    op[1] = make_float4(yo[4], yo[5], yo[6], yo[7]);   // row written contiguously
  }
}

// ---------------- launcher ----------------

extern "C" void kernel_launch(void* const* d_in, const int* in_sizes, int n_in,
                              void* d_out, int out_size, void* d_ws, size_t ws_size,
                              hipStream_t stream) {
  const float* h     = (const float*)d_in[0];
  const int*   ei    = (const int*)  d_in[1];
  const float* ew    = (const float*)d_in[2];
  const float* Wm    = (const float*)d_in[3];
  const float* bias  = (const float*)d_in[4];
  const float* gamma = (const float*)d_in[5];
  const float* beta  = (const float*)d_in[6];

  int n = in_sizes[0] / C;
  int E = in_sizes[2];
  const int* src = ei;
  const int* dst = ei + E;

  // workspace layout: agg[N*C] | deg[N] | d[N]   (~52 MB)
  float* agg = (float*)d_ws;
  float* deg = agg + (size_t)n * C;
  float* dnv = deg + n;
  float* out = (float*)d_out;

  if (n <= 0) return;

  k_init_deg<<<(n + 255) / 256, 256, 0, stream>>>(deg, n);
  k_deg_acc <<<(E + 255) / 256, 256, 0, stream>>>(dst, ew, deg, E);
  k_dinv    <<<(n + 255) / 256, 256, 0, stream>>>(deg, dnv, n);
  k_self    <<<(n * 32 + 255) / 256, 256, 0, stream>>>(h, dnv, agg, n);
  k_scatter <<<(E + 7) / 8, 256, 0, stream>>>(h, dnv, src, dst, ew, agg, E);
  k_gemm_ln_gelu<<<(n + 15) / 16, 256, 0, stream>>>(agg, h, Wm, bias, gamma, beta, out, n);
}